// PrototypeLayer_37177236914458
// MI455X (gfx1250) — compile-verified
//
#include <hip/hip_runtime.h>

typedef __attribute__((ext_vector_type(2))) float        v2f;
typedef __attribute__((ext_vector_type(8))) float        v8f;
typedef __attribute__((ext_vector_type(4))) unsigned int v4u;
typedef __attribute__((ext_vector_type(4))) int          v4i;
typedef __attribute__((ext_vector_type(8))) int          v8i;

#define B_DIM 32
#define T_DIM 64
#define F_DIM 32
#define C_DIM 64
#define P_DIM 32
#define ROW_STR (T_DIM * F_DIM * C_DIM)  // 131072 floats per b (or p) row
#define T_STR   (F_DIM * C_DIM)          // 2048 floats per t step

#define T_CH    4                        // t-values per TDM chunk
#define NCHUNK  (T_DIM / T_CH)           // 16 chunks
#define SLABF   (B_DIM * T_CH * C_DIM)   // 8192 floats per slab (32 KB)

// TDM: 3-D tile (x=c:64 contiguous, y=t:T_CH stride 2048, z=b:32 stride 131072)
// DMA'd from global into a dense LDS slab laid out [b][t_local][c].
__device__ __forceinline__ void tdm_load_slab(const float* gptr, unsigned lds_byte) {
    unsigned long long ga = (unsigned long long)(uintptr_t)gptr;
    v4u g0;
    g0[0] = 1u;                                   // count=1 (valid), user mode
    g0[1] = lds_byte;                             // lds_addr
    g0[2] = (unsigned)(ga & 0xFFFFFFFFu);         // global_addr[31:0]
    g0[3] = (unsigned)((ga >> 32) & 0x01FFFFFFu)  // global_addr[56:32]
            | (2u << 30);                         // type = 2 ("image")
    v8i g1;
    g1[0] = (2 << 16);                            // data_size = 4B; no mask/pad/iter
    g1[1] = (C_DIM << 16);                        // tensor_dim0[15:0] = 64
    g1[2] = (T_DIM << 16);                        // tensor_dim1[15:0] = 64
    g1[3] = (C_DIM << 16);                        // tile_dim0 = 64
    g1[4] = T_CH | (B_DIM << 16);                 // tile_dim1 = 4, tile_dim2 = 32
    g1[5] = T_STR;                                // tensor_dim0_stride = 2048
    g1[6] = 0;                                    // stride0 hi / stride1 lo16 = 0
    g1[7] = (int)(ROW_STR >> 16);                 // tensor_dim1_stride = 131072
    v4i g2;
    g2[0] = B_DIM;                                // tensor_dim2 = 32
    g2[1] = 0; g2[2] = 0; g2[3] = 0;              // dim3 / stride2 / tile_dim3 unused
    v4i g3 = {0, 0, 0, 0};
#if defined(__clang_major__) && __clang_major__ >= 23
    v8i g4 = {0, 0, 0, 0, 0, 0, 0, 0};
    __builtin_amdgcn_tensor_load_to_lds(g0, g1, g2, g3, g4, 0);
#else
    __builtin_amdgcn_tensor_load_to_lds(g0, g1, g2, g3, 0);
#endif
}

// One block per f. 16 waves: wave = ks*4 + tile, tile = tile_p*2 + tile_b.
// TDM double-buffers (x, kernel) slabs in LDS; each wave owns t_local = ks.
__global__ __launch_bounds__(512) void proto_dist_kernel(
    const float* __restrict__ x, const float* __restrict__ kern,
    float* __restrict__ out)
{
    const int f       = blockIdx.x;
    const int lane    = threadIdx.x & 31;
    const int wave    = threadIdx.x >> 5;   // 0..15
    const int tile    = wave & 3;
    const int ks      = wave >> 2;          // 0..3 (t_local within chunk)
    const int tile_b  = tile & 1;
    const int tile_p  = tile >> 1;
    const int lane_lo = lane & 15;
    const int lane_hi = lane >> 4;

    extern __shared__ float smem[];
    float* xs   = smem;                 // [2][SLABF] x slabs
    float* ksl  = smem + 2 * SLABF;     // [2][SLABF] kernel slabs
    float* cbuf = smem + 4 * SLABF;     // [16][256] partial C tiles
    float* xsp  = cbuf + 16 * 256;      // [16][32]
    float* ksp  = xsp + 16 * 32;        // [16][32]
    float* xsqf = ksp + 16 * 32;        // [32]
    float* ksqf = xsqf + 32;            // [32]

    const int b_row = tile_b * 16 + lane_lo;
    const int p_row = tile_p * 16 + lane_lo;

    // Prefetch chunks 0 and 1 (2 TDM ops per chunk, in-order completion).
    if (wave == 0) {
        tdm_load_slab(x    + 0 * T_CH * T_STR + f * C_DIM, (unsigned)(uintptr_t)(xs));
        tdm_load_slab(kern + 0 * T_CH * T_STR + f * C_DIM, (unsigned)(uintptr_t)(ksl));
        tdm_load_slab(x    + 1 * T_CH * T_STR + f * C_DIM, (unsigned)(uintptr_t)(xs + SLABF));
        tdm_load_slab(kern + 1 * T_CH * T_STR + f * C_DIM, (unsigned)(uintptr_t)(ksl + SLABF));
    }

    v8f acc = {};
    v2f axv = {0.f, 0.f}, bxv = {0.f, 0.f};

    for (int i = 0; i < NCHUNK; ++i) {
        if (wave == 0) {
            if (i < NCHUNK - 1) __builtin_amdgcn_s_wait_tensorcnt(2);
            else                __builtin_amdgcn_s_wait_tensorcnt(0);
        }
        __syncthreads();   // chunk i resident for all waves

        const float* xrow = xs  + (i & 1) * SLABF + (b_row * T_CH + ks) * C_DIM + lane_hi * 2;
        const float* krow = ksl + (i & 1) * SLABF + (p_row * T_CH + ks) * C_DIM + lane_hi * 2;
        #pragma unroll
        for (int c = 0; c < C_DIM; c += 4) {
            v2f av = *(const v2f*)(xrow + c);
            v2f bv = *(const v2f*)(krow + c);
            acc = __builtin_amdgcn_wmma_f32_16x16x4_f32(
                false, av, false, bv, (short)0, acc, false, false);
            axv += av * av;           // single v_pk_fma_f32 each
            bxv += bv * bv;
        }

        __syncthreads();   // buffer (i&1) free for reuse
        if (wave == 0 && i + 2 < NCHUNK) {
            const int j = i + 2;
            tdm_load_slab(x    + j * T_CH * T_STR + f * C_DIM,
                          (unsigned)(uintptr_t)(xs  + (j & 1) * SLABF));
            tdm_load_slab(kern + j * T_CH * T_STR + f * C_DIM,
                          (unsigned)(uintptr_t)(ksl + (j & 1) * SLABF));
        }
    }

    #pragma unroll
    for (int r = 0; r < 8; ++r)
        cbuf[wave * 256 + r * 32 + lane] = acc[r];
    xsp[wave * 32 + lane] = axv.x + axv.y;
    ksp[wave * 32 + lane] = bxv.x + bxv.y;
    __syncthreads();

    // Full row norms (count each K once: tile_p==0 waves for x, tile_b==0 for k).
    if (wave == 0) {
        const int row = lane;
        float s = 0.f;
        #pragma unroll
        for (int k = 0; k < 4; ++k) {
            const int w = k * 4 + (row >> 4);          // tile_p == 0
            s += xsp[w * 32 + (row & 15)] + xsp[w * 32 + (row & 15) + 16];
        }
        xsqf[row] = s;
    } else if (wave == 1) {
        const int row = lane;
        float s = 0.f;
        #pragma unroll
        for (int k = 0; k < 4; ++k) {
            const int w = k * 4 + 2 * (row >> 4);      // tile_b == 0
            s += ksp[w * 32 + (row & 15)] + ksp[w * 32 + (row & 15) + 16];
        }
        ksqf[row] = s;
    }
    __syncthreads();

    // Leader waves (ks==0) reduce K-split partials and fuse the norm terms.
    if (ks == 0) {
        #pragma unroll
        for (int r = 0; r < 8; ++r) {
            float csum = 0.f;
            #pragma unroll
            for (int k = 0; k < 4; ++k)
                csum += cbuf[(k * 4 + tile) * 256 + r * 32 + lane];
            // C/D layout: VGPR r, lanes 0-15 -> M=r, lanes 16-31 -> M=r+8.
            const int b = tile_b * 16 + r + lane_hi * 8;
            const int p = tile_p * 16 + lane_lo;
            out[b * (P_DIM * F_DIM) + p * F_DIM + f] =
                xsqf[b] - 2.0f * csum + ksqf[p];
        }
    }
}

extern "C" void kernel_launch(void* const* d_in, const int* in_sizes, int n_in,
                              void* d_out, int out_size, void* d_ws, size_t ws_size,
                              hipStream_t stream) {
    const float* x    = (const float*)d_in[0];  // (32, 64, 32, 64) f32
    const float* kern = (const float*)d_in[1];  // (32, 64, 32, 64) f32
    float* out = (float*)d_out;                 // (32, 32, 32) f32
    (void)in_sizes; (void)n_in; (void)out_size; (void)d_ws; (void)ws_size;
    const size_t shmem = (4 * SLABF + 16 * 256 + 2 * 16 * 32 + 64) * sizeof(float);
    proto_dist_kernel<<<dim3(F_DIM), dim3(512), shmem, stream>>>(x, kern, out);
}